// GCN_dep_44951127720457
// MI455X (gfx1250) — compile-verified
//
#include <hip/hip_runtime.h>

#define NHID 128
#define NCLS 10

typedef float v2f __attribute__((ext_vector_type(2)));
typedef float v8f __attribute__((ext_vector_type(8)));

// ---------------------------------------------------------------------------
// C[M x 128] = A[M x K] * W[K x 128]  (fp32-exact via V_WMMA_F32_16X16X4_F32)
// Block: 256 threads = 8 waves; each wave -> one 16x16 output tile.
// Block covers 128 rows x 16 cols. gridDim = (ceil(M/128), 128/16).
// ---------------------------------------------------------------------------
__global__ __launch_bounds__(256) void gemm_f32_wmma(
    const float* __restrict__ A, const float* __restrict__ W,
    float* __restrict__ C, int M, int K)
{
    __shared__ float Wl[256 * 16];          // K_max(=256) x 16 column tile of W
    const int n0 = blockIdx.y * 16;

    // Cooperative stage of the K x 16 weight slice into LDS (reused by 8 waves)
    for (int idx = threadIdx.x; idx < K * 16; idx += 256) {
        const int k = idx >> 4, n = idx & 15;
        Wl[idx] = W[k * NHID + n0 + n];
    }
    __syncthreads();

    const int wave = threadIdx.x >> 5;
    const int lane = threadIdx.x & 31;
    const int half = lane >> 4;             // 0: K offset 0..1, 1: K offset 2..3
    const int nl   = lane & 15;
    const int r0   = blockIdx.x * 128 + wave * 16;

    // Clamp OOB rows for loads (EXEC must stay all-ones around WMMA);
    // stores are predicated instead.
    int mrow = r0 + nl;
    if (mrow > M - 1) mrow = M - 1;
    const float* aptr = A + (size_t)mrow * K;

    v8f acc = {};
    for (int k = 0; k < K; k += 4) {
        const int kb = k + half * 2;
        // A 16x4 fp32 layout: lane m=nl holds K=kb, kb+1 in 2 VGPRs
        v2f a = *(const v2f*)(aptr + kb);
        // B 4x16 fp32 layout (mirror of A): lane n=nl holds K=kb, kb+1
        v2f b;
        b.x = Wl[kb * 16 + nl];
        b.y = Wl[(kb + 1) * 16 + nl];
        // (neg_a, A, neg_b, B, c_mod, C, reuse_a, reuse_b)
        acc = __builtin_amdgcn_wmma_f32_16x16x4_f32(false, a, false, b,
                                                    (short)0, acc, false, false);
    }

    // C/D layout: VGPR r -> row r (lanes 0-15) / row r+8 (lanes 16-31), N = nl
#pragma unroll
    for (int r = 0; r < 8; ++r) {
        const int row = r0 + half * 8 + r;
        if (row < M) C[(size_t)row * NHID + n0 + nl] = acc[r];
    }
}

// ---------------------------------------------------------------------------
// SpMM scatter: one wave per edge; lanes cover all 128 features as float4.
// Gather = coalesced 512B read (L2-resident), scatter = f32 global atomics.
// ---------------------------------------------------------------------------
__global__ __launch_bounds__(256) void spmm_scatter(
    const float* __restrict__ S, const int* __restrict__ src,
    const int* __restrict__ dst, const float* __restrict__ w,
    float* __restrict__ agg, int E)
{
    const long long gid = (long long)blockIdx.x * blockDim.x + threadIdx.x;
    const int edge = (int)(gid >> 5);
    const int lane = (int)(gid & 31);
    if (edge >= E) return;

    const int   s  = src[edge];       // wave-uniform
    const int   d  = dst[edge];       // wave-uniform
    const float we = w[edge];         // wave-uniform

    const float4 v = *(const float4*)(S + (size_t)s * NHID + lane * 4);
    float* p = agg + (size_t)d * NHID + lane * 4;
    atomicAdd(p + 0, v.x * we);
    atomicAdd(p + 1, v.y * we);
    atomicAdd(p + 2, v.z * we);
    atomicAdd(p + 3, v.w * we);
}

// ---------------------------------------------------------------------------
__global__ void fill_zero4(float* __restrict__ p, size_t n)
{
    const size_t i = ((size_t)blockIdx.x * blockDim.x + threadIdx.x) * 4;
    if (i < n) *(float4*)(p + i) = make_float4(0.f, 0.f, 0.f, 0.f);
}

__global__ void bias_act(const float* __restrict__ a, const float* __restrict__ b,
                         float* __restrict__ h, size_t n)
{
    const size_t i = (size_t)blockIdx.x * blockDim.x + threadIdx.x;
    if (i >= n) return;
    const float v = a[i] + b[i & (NHID - 1)];
    h[i] = fmaxf(v, 0.f);
}

// 128 threads per block; each block sums 256 rows of its column, then atomics.
__global__ void colsum_kernel(const float* __restrict__ agg,
                              float* __restrict__ sums, int M)
{
    const int col = threadIdx.x;
    const int r0 = blockIdx.x * 256;
    const int rend = min(r0 + 256, M);
    float acc = 0.f;
    for (int r = r0; r < rend; ++r) acc += agg[(size_t)r * NHID + col];
    atomicAdd(&sums[col], acc);
}

// g = colsum/N + b3 ; out = g @ lin_w + lin_b   (lanes 0..9 -> one class each)
__global__ void final_linear(const float* __restrict__ csum,
                             const float* __restrict__ b3,
                             const float* __restrict__ lw,
                             const float* __restrict__ lb,
                             float* __restrict__ out, float invN)
{
    const int c = threadIdx.x;
    if (c >= NCLS) return;
    float acc = lb[c];
    for (int k = 0; k < NHID; ++k) {
        const float g = csum[k] * invN + b3[k];
        acc += g * lw[k * NCLS + c];
    }
    out[c] = acc;
}

// ---------------------------------------------------------------------------
extern "C" void kernel_launch(void* const* d_in, const int* in_sizes, int n_in,
                              void* d_out, int out_size, void* d_ws, size_t ws_size,
                              hipStream_t stream)
{
    const float* x  = (const float*)d_in[0];
    const int*   ei = (const int*)  d_in[1];
    const float* ew = (const float*)d_in[2];
    const float* W1 = (const float*)d_in[3];
    const float* b1 = (const float*)d_in[4];
    const float* W2 = (const float*)d_in[5];
    const float* b2 = (const float*)d_in[6];
    const float* W3 = (const float*)d_in[7];
    const float* b3 = (const float*)d_in[8];
    const float* lw = (const float*)d_in[9];
    const float* lb = (const float*)d_in[10];
    float* out = (float*)d_out;

    const int NF = 256;
    const int N  = in_sizes[0] / NF;     // 100000 nodes
    const int E  = in_sizes[2];          // 1600000 edges
    const int* src = ei;
    const int* dst = ei + E;

    char* ws = (char*)d_ws;
    const size_t bufBytes = (size_t)N * NHID * sizeof(float);   // 51.2 MB
    float* buf0 = (float*)ws;
    float* buf1 = (float*)(ws + bufBytes);
    float* csum = (float*)(ws + 2 * bufBytes);

    const size_t nh = (size_t)N * NHID;                          // 12.8M
    const dim3 gemmGrid((N + 127) / 128, NHID / 16);
    const int spmmBlocks = (int)(((long long)E * 32 + 255) / 256);
    const int zeroBlocks = (int)((nh / 4 + 255) / 256);
    const int eltBlocks  = (int)((nh + 255) / 256);

    // ---- layer 1: support1 = x @ W1 ; agg1 = scatter ; h1 = relu(agg1+b1)
    gemm_f32_wmma<<<gemmGrid, 256, 0, stream>>>(x, W1, buf0, N, NF);
    fill_zero4<<<zeroBlocks, 256, 0, stream>>>(buf1, nh);
    spmm_scatter<<<spmmBlocks, 256, 0, stream>>>(buf0, src, dst, ew, buf1, E);
    bias_act<<<eltBlocks, 256, 0, stream>>>(buf1, b1, buf0, nh);

    // ---- layer 2
    gemm_f32_wmma<<<gemmGrid, 256, 0, stream>>>(buf0, W2, buf1, N, NHID);
    fill_zero4<<<zeroBlocks, 256, 0, stream>>>(buf0, nh);
    spmm_scatter<<<spmmBlocks, 256, 0, stream>>>(buf1, src, dst, ew, buf0, E);
    bias_act<<<eltBlocks, 256, 0, stream>>>(buf0, b2, buf1, nh);

    // ---- layer 3 (no relu; bias folded into the pooled mean)
    gemm_f32_wmma<<<gemmGrid, 256, 0, stream>>>(buf1, W3, buf0, N, NHID);
    fill_zero4<<<zeroBlocks, 256, 0, stream>>>(buf1, nh);
    spmm_scatter<<<spmmBlocks, 256, 0, stream>>>(buf0, src, dst, ew, buf1, E);

    // ---- mean pool + classifier
    fill_zero4<<<1, 32, 0, stream>>>(csum, (size_t)NHID);
    colsum_kernel<<<(N + 255) / 256, NHID, 0, stream>>>(buf1, csum, N);
    final_linear<<<1, 32, 0, stream>>>(csum, b3, lw, lb, out, 1.0f / (float)N);
}